// MSTParserLSTM_83124797047273
// MI455X (gfx1250) — compile-verified
//
#include <hip/hip_runtime.h>

// Problem constants (from reference)
#define B_    16
#define N_    1024
#define D2H_  1024
#define U_    512

typedef float v2f __attribute__((ext_vector_type(2)));
typedef float v8f __attribute__((ext_vector_type(8)));

// Workspace layout (float offsets)
#define WS_VH  0                      // v_h[1024] = W_h^T w_out
#define WS_VM  1024                   // v_m[1024] = W_m^T w_out  (contiguous after VH)
#define WS_C   2048                   // [0] = b_h.w_out + b_out, [1] = b_m.w_out
#define WS_SH  2064                   // sH[16384]
#define WS_SM  (2064 + (B_ * N_))     // sM[16384]

// ---------------------------------------------------------------------------
// Kernel A: fold weight matrices into vectors: v[d] = sum_u w_out[u] * W[u][d]
// Coalesced across d (consecutive threads -> consecutive columns).
// ---------------------------------------------------------------------------
__global__ __launch_bounds__(256) void fold_weights(const float* __restrict__ Wh,
                                                    const float* __restrict__ Wm,
                                                    const float* __restrict__ wout,
                                                    float* __restrict__ ws) {
  const int d = blockIdx.x * blockDim.x + threadIdx.x;   // 0..1023
  float vh = 0.f, vm = 0.f;
  for (int u = 0; u < U_; ++u) {
    const float w = wout[u];                             // uniform -> scalar load
    vh = fmaf(w, Wh[u * D2H_ + d], vh);
    vm = fmaf(w, Wm[u * D2H_ + d], vm);
  }
  ws[WS_VH + d] = vh;
  ws[WS_VM + d] = vm;
}

// ---------------------------------------------------------------------------
// Kernel A2: scalar folds  cH = w_out.b_h + b_out ;  cM = w_out.b_m
// ---------------------------------------------------------------------------
__global__ __launch_bounds__(512) void fold_bias(const float* __restrict__ bh,
                                                 const float* __restrict__ bm,
                                                 const float* __restrict__ wout,
                                                 const float* __restrict__ bout,
                                                 float* __restrict__ ws) {
  __shared__ float sh[512];
  __shared__ float sm[512];
  const int t = threadIdx.x;
  const float w = wout[t];
  sh[t] = w * bh[t];
  sm[t] = w * bm[t];
  __syncthreads();
  for (int s = 256; s > 0; s >>= 1) {
    if (t < s) { sh[t] += sh[t + s]; sm[t] += sm[t + s]; }
    __syncthreads();
  }
  if (t == 0) {
    ws[WS_C + 0] = sh[0] + bout[0];
    ws[WS_C + 1] = sm[0];
  }
}

// ---------------------------------------------------------------------------
// Kernel B: sH/sM = x . {v_h, v_m} via V_WMMA_F32_16X16X4_F32.
// Block = 256 threads = 8 waves; block covers 128 rows of x (16 rows / wave).
// x is staged through LDS in KC=64 chunks with coalesced float4 loads;
// row stride padded to 68 floats so the strided A-matrix reads are
// bank-conflict free (bank step of 4 per M).
// ---------------------------------------------------------------------------
#define ROWS_PB 128
#define KC      64
#define LDST    68   // padded LDS row stride in floats

__global__ __launch_bounds__(256) void gemv_wmma(const float* __restrict__ x,
                                                 float* __restrict__ ws) {
  __shared__ float ldsX[ROWS_PB * LDST];   // ~34 KB
  __shared__ float ldsV[2 * D2H_];         //  8 KB : v_h then v_m

  const int t    = threadIdx.x;
  const int lane = t & 31;
  const int w    = t >> 5;                 // wave id 0..7
  const int row0 = blockIdx.x * ROWS_PB;

  // Stage v_h / v_m (contiguous in ws) into LDS once.
  for (int i = t; i < 2 * D2H_; i += 256) ldsV[i] = ws[WS_VH + i];

  const int mrow = lane & 15;              // A-matrix row within tile / B col N
  const int hi   = lane >> 4;              // 0: K=k,k+1   1: K=k+2,k+3

  v8f acc = {};                            // 16x16 f32 accumulator (cols 0,1 used)

  for (int kc = 0; kc < D2H_; kc += KC) {
    __syncthreads();                       // LDS reuse guard (incl. ldsV stage)

    // Cooperative load: 128 rows x 64 floats = 2048 float4, 8 per thread.
    #pragma unroll
    for (int i = 0; i < 8; ++i) {
      const int idx = i * 256 + t;         // 0..2047
      const int r   = idx >> 4;            // 16 float4 per row
      const int c4  = (idx & 15) << 2;     // float offset within chunk
      const float4 v = *(const float4*)(x + (size_t)(row0 + r) * D2H_ + kc + c4);
      *(float4*)&ldsX[r * LDST + c4] = v;
    }

    // Prefetch the next K-chunk (gfx1250 global_prefetch_b8).
    if (kc + KC < D2H_) {
      __builtin_prefetch(x + (size_t)(row0 + (t >> 1)) * D2H_ + (kc + KC) + ((t & 1) << 5), 0, 0);
    }
    __syncthreads();

    // Wave w: 16-row tile, K swept 4 at a time through the f32 WMMA.
    const float* ar = &ldsX[(w * 16 + mrow) * LDST];
    #pragma unroll 4
    for (int k = 0; k < KC; k += 4) {
      // A 16x4: lane=M; VGPR pair holds (K=k+2*hi, K=k+2*hi+1)
      v2f a = *(const v2f*)(ar + k + 2 * hi);
      // B 4x16: lane=N; col 0 = v_h, col 1 = v_m, cols 2..15 = 0
      v2f b = {0.f, 0.f};
      if (mrow < 2) {
        const float* vp = &ldsV[mrow * D2H_ + kc + k + 2 * hi];
        b.x = vp[0];
        b.y = vp[1];
      }
      acc = __builtin_amdgcn_wmma_f32_16x16x4_f32(
          /*neg_a=*/false, a, /*neg_b=*/false, b,
          /*c_mod=*/(short)0, acc, /*reuse_a=*/false, /*reuse_b=*/false);
    }
  }

  // D layout: VGPR r = rows {r, r+8}, N = lane%16.  Col 0 -> sH, col 1 -> sM.
  if (mrow < 2) {
    const float c = ws[WS_C + mrow];       // fold bias scalars here
    float* dst = ws + (mrow == 0 ? WS_SH : WS_SM) + row0 + w * 16 + hi * 8;
    #pragma unroll
    for (int r = 0; r < 8; ++r) dst[r] = acc[r] + c;
  }
}

// ---------------------------------------------------------------------------
// Kernel C: scores[b,i,j] = sH[b,i] + sM[b,j]   (64 MB, pure streaming write)
// One block per (b,i) row; float4 stores; sM row hits L2/WGP$ across i.
// ---------------------------------------------------------------------------
__global__ __launch_bounds__(256) void expand_scores(const float* __restrict__ ws,
                                                     float* __restrict__ out) {
  const int bi = blockIdx.x;               // b*N + i
  const int b  = bi >> 10;
  const float s_i = ws[WS_SH + bi];
  const int j4 = threadIdx.x << 2;
  const float4 m = *(const float4*)(ws + WS_SM + (b << 10) + j4);
  float4 r;
  r.x = s_i + m.x; r.y = s_i + m.y; r.z = s_i + m.z; r.w = s_i + m.w;
  *(float4*)(out + (size_t)bi * N_ + j4) = r;
}

// ---------------------------------------------------------------------------
extern "C" void kernel_launch(void* const* d_in, const int* in_sizes, int n_in,
                              void* d_out, int out_size, void* d_ws, size_t ws_size,
                              hipStream_t stream) {
  const float* x   = (const float*)d_in[0];   // [B,N,D2H]
  const float* Wh  = (const float*)d_in[1];   // [U,D2H]
  const float* bh  = (const float*)d_in[2];   // [U]
  const float* Wm  = (const float*)d_in[3];   // [U,D2H]
  const float* bm  = (const float*)d_in[4];   // [U]
  const float* wo  = (const float*)d_in[5];   // [U]
  const float* bo  = (const float*)d_in[6];   // [1]
  float* out = (float*)d_out;                 // [B,N,N]
  float* ws  = (float*)d_ws;

  fold_weights<<<D2H_ / 256, 256, 0, stream>>>(Wh, Wm, wo, ws);
  fold_bias<<<1, 512, 0, stream>>>(bh, bm, wo, bo, ws);
  gemv_wmma<<<(B_ * N_) / ROWS_PB, 256, 0, stream>>>(x, ws);
  expand_scores<<<B_ * N_, 256, 0, stream>>>(ws, out);
}